// MultiScale_Topology_Kernel_31817117728895
// MI455X (gfx1250) — compile-verified
//
#include <hip/hip_runtime.h>
#include <hip/hip_bf16.h>
#include <math.h>

// ---------------- problem constants ----------------
#define NATOMS 1536
#define NGRP   192
#define LB     24
#define GB     24
#define HDIM   128
#define CUTR   12.0f
#define ALPHA_ 0.18f

// ---------------- output layout (floats) ----------------
#define OUT_LOCAL  ((size_t)0)
#define OUT_GLOBAL ((size_t)56623104)          // N*N*24
#define OUT_ATOMIC ((size_t)113246208)         // + N*N*24
#define OUT_GROUP  ((size_t)113442816)         // + N*128
#define OUT_CONF   ((size_t)113467392)         // + G*128
#define OUT_DIST   ((size_t)113467520)         // + 128

// ---------------- workspace layout (floats) ----------------
#define WS_H       ((size_t)0)                 // N*52          = 79872
#define WS_MID1    ((size_t)79872)             // N*128         = 196608
#define WS_GRP_ACC ((size_t)276480)            // G*176 = 33792 [0..127 atomic | 128..151 lp | 152..175 gp]
#define WS_GRP_IN  ((size_t)310272)            // G*176         = 33792
#define WS_GMID    ((size_t)344064)            // G*128         = 24576
#define WS_COUNTS  ((size_t)368640)            // G             = 192
#define WS_ASUM    ((size_t)368832)            // 128
#define WS_GPSUM   ((size_t)368960)            // 24
#define WS_PSUM    ((size_t)368984)            // 3
#define WS_FSUM    ((size_t)368987)            // 3
#define WS_END     ((size_t)368990)

typedef float v2f __attribute__((ext_vector_type(2)));
typedef float v4f __attribute__((ext_vector_type(4)));
typedef float v8f __attribute__((ext_vector_type(8)));

// =====================================================================
// Kernel 1: pairwise dist + local/global RBF streaming + pooled sums
// one block (256 threads, 8 waves) per atom row i
// 463 MB of dead-on-write output -> non-temporal B128 stores
// =====================================================================
__global__ void rbf_kernel(const float* __restrict__ pos,
                           const float* __restrict__ species,
                           const float* __restrict__ forces,
                           const int*   __restrict__ assign,
                           const float* __restrict__ lc,
                           const float* __restrict__ gc,
                           float* __restrict__ out_local,
                           float* __restrict__ out_global,
                           float* __restrict__ out_dist,
                           float* __restrict__ ws_h,
                           float* __restrict__ grp_acc,
                           float* __restrict__ counts,
                           float* __restrict__ gp_sum,
                           float* __restrict__ pos_sum,
                           float* __restrict__ f_sum)
{
    __shared__ float s_lc[LB];
    __shared__ float s_gc[GB];
    __shared__ float s_red[8][48];
    __shared__ float s_p[3];

    const int i   = blockIdx.x;
    const int tid = threadIdx.x;

    if (tid < LB)              s_lc[tid]      = lc[tid];
    else if (tid < LB + GB)    s_gc[tid - LB] = gc[tid - LB];
    if (tid < 3)               s_p[tid]       = pos[i * 3 + tid];
    __syncthreads();

    const float px = s_p[0], py = s_p[1], pz = s_p[2];
    const float width = (2.5f - 1.0f) / (float)(LB - 1);
    const float gamma = 1.0f / fmaxf(width * width, 1e-6f);

    float lp[LB], gp[GB];
#pragma unroll
    for (int b = 0; b < LB; ++b) { lp[b] = 0.f; gp[b] = 0.f; }

#pragma unroll 1
    for (int jj = 0; jj < NATOMS / 256; ++jj) {
        const int j = tid + jj * 256;
        const float dx = px - pos[j * 3 + 0];
        const float dy = py - pos[j * 3 + 1];
        const float dz = pz - pos[j * 3 + 2];
        float r2 = dx * dx + dy * dy + dz * dz;
        r2 = fmaxf(r2, 1e-8f);
        const float d = sqrtf(r2);
        __builtin_nontemporal_store(d, out_dist + (size_t)i * NATOMS + j);

        const bool  ne    = (i != j);
        const float lflag = (ne && d >= 1.0f && d <= 2.5f) ? 1.0f : 0.0f;
        const float gflag = (ne && d >  2.5f && d <= CUTR) ? 1.0f : 0.0f;

        float* __restrict__ ol = out_local  + ((size_t)i * NATOMS + j) * LB;
        float* __restrict__ og = out_global + ((size_t)i * NATOMS + j) * GB;

        const float scr = erfcf(0.35f * d);
        const float q   = d * (1.0f / CUTR);
        const float tail = 0.2f / (1.0f + q * q);

#pragma unroll
        for (int b4 = 0; b4 < LB; b4 += 4) {
            v4f lv4, gv4;
#pragma unroll
            for (int u = 0; u < 4; ++u) {
                const int   b  = b4 + u;
                const float t1 = d - s_lc[b];
                const float lv = lflag * expf(-gamma * t1 * t1);
                lv4[u] = lv;
                lp[b] += lv;
                const float t2 = d - s_gc[b];
                const float gv = gflag * (expf(-ALPHA_ * t2 * t2) * scr + tail);
                gv4[u] = gv;
                gp[b] += gv;
            }
            __builtin_nontemporal_store(lv4, (v4f*)(ol + b4));   // global_store_b128 NT
            __builtin_nontemporal_store(gv4, (v4f*)(og + b4));   // global_store_b128 NT
        }
    }

    // wave32 butterfly reduction of the 48 pooled sums
#pragma unroll
    for (int b = 0; b < LB; ++b) {
#pragma unroll
        for (int off = 16; off > 0; off >>= 1) {
            lp[b] += __shfl_xor(lp[b], off, 32);
            gp[b] += __shfl_xor(gp[b], off, 32);
        }
    }
    const int wv = tid >> 5, ln = tid & 31;
    if (ln == 0) {
#pragma unroll
        for (int b = 0; b < LB; ++b) {
            s_red[wv][b]      = lp[b];
            s_red[wv][LB + b] = gp[b];
        }
    }
    __syncthreads();

    if (tid < 48) {
        float s = 0.f;
#pragma unroll
        for (int w = 0; w < 8; ++w) s += s_red[w][tid];
        const int g = assign[i];
        ws_h[(size_t)i * 52 + tid] = s;
        if (tid < LB) {
            atomicAdd(&grp_acc[(size_t)g * 176 + 128 + tid], s);
        } else {
            const int b = tid - LB;
            atomicAdd(&grp_acc[(size_t)g * 176 + 152 + b], s);
            atomicAdd(&gp_sum[b], s);
        }
    } else if (tid < 51) {
        const int c = tid - 48;
        ws_h[(size_t)i * 52 + tid] = s_p[c];
        atomicAdd(&pos_sum[c], s_p[c]);
    } else if (tid == 51) {
        ws_h[(size_t)i * 52 + 51] = species[i] * (1.0f / 60.0f);
    } else if (tid < 55) {
        const int c = tid - 52;
        atomicAdd(&f_sum[c], forces[i * 3 + c]);
    } else if (tid == 55) {
        atomicAdd(&counts[assign[i]], 1.0f);
    }
}

// =====================================================================
// WMMA f32 16x16x4 GEMM: C = act(A[M,K] @ B[K,Nc] + bias)
// one wave per 16x16 output tile; blockDim = 256 (8 waves)
// mode 0: SiLU   mode 1: plain   mode 2: plain + segment/mean atomics
// =====================================================================
__global__ void gemm_wmma_kernel(const float* __restrict__ A,
                                 const float* __restrict__ B,
                                 const float* __restrict__ bias,
                                 float* __restrict__ C,
                                 int M, int Nc, int K, int mode,
                                 const int* __restrict__ assign,
                                 float* __restrict__ grp_acc,
                                 float* __restrict__ vec_acc)
{
    const int wave = threadIdx.x >> 5;
    const int lane = threadIdx.x & 31;
    const int tile = blockIdx.x * (blockDim.x >> 5) + wave;
    const int ntiles = Nc >> 4;
    const int m0 = (tile / ntiles) << 4;
    const int n0 = (tile % ntiles) << 4;
    const int lh  = lane >> 4;   // lane half selects K+0..1 vs K+2..3
    const int l16 = lane & 15;

    v8f acc = {0.f, 0.f, 0.f, 0.f, 0.f, 0.f, 0.f, 0.f};
    const int arow = m0 + l16;

    for (int k = 0; k < K; k += 4) {
        v2f a, b;
        const float* ap = A + (size_t)arow * K + k + 2 * lh;   // A: M x K row-major
        a.x = ap[0];
        a.y = ap[1];
        const int kb = k + 2 * lh;                              // B: K x Nc row-major
        b.x = B[(size_t)kb * Nc + n0 + l16];
        b.y = B[(size_t)(kb + 1) * Nc + n0 + l16];
        acc = __builtin_amdgcn_wmma_f32_16x16x4_f32(
            /*neg_a=*/false, a, /*neg_b=*/false, b,
            /*c_mod=*/(short)0, acc, /*reuse_a=*/false, /*reuse_b=*/false);
    }

    const int col = n0 + l16;
    const float bv = bias[col];
#pragma unroll
    for (int v = 0; v < 8; ++v) {
        const int row = m0 + v + 8 * lh;
        float val = acc[v] + bv;
        if (mode == 0) val = val / (1.0f + expf(-val));   // SiLU
        C[(size_t)row * Nc + col] = val;
        if (mode == 2) {
            const int g = assign[row];
            atomicAdd(&grp_acc[(size_t)g * 176 + col], val);
            atomicAdd(&vec_acc[col], val);
        }
    }
}

// =====================================================================
// grp_in[g, 0:176] = grp_acc[g, :] / counts[g]
// =====================================================================
__global__ void grp_finalize(const float* __restrict__ grp_acc,
                             const float* __restrict__ counts,
                             float* __restrict__ grp_in)
{
    const int idx = blockIdx.x * blockDim.x + threadIdx.x;
    if (idx >= NGRP * 176) return;
    grp_in[idx] = grp_acc[idx] / counts[idx / 176];
}

// =====================================================================
// conformer MLP (M=1): plain dot products
// =====================================================================
__global__ void conformer_kernel(const float* __restrict__ asum,
                                 const float* __restrict__ gpsum,
                                 const float* __restrict__ psum,
                                 const float* __restrict__ fsum,
                                 const float* __restrict__ w1,
                                 const float* __restrict__ b1,
                                 const float* __restrict__ w2,
                                 const float* __restrict__ b2,
                                 float* __restrict__ out)
{
    __shared__ float cin[158];
    __shared__ float mid[HDIM];
    const int t = threadIdx.x;
    const float invN = 1.0f / (float)NATOMS;
    if (t < 128)      cin[t] = asum[t] * invN;
    else if (t < 152) cin[t] = gpsum[t - 128] * invN;
    else if (t < 155) cin[t] = psum[t - 152] * invN;
    else if (t < 158) cin[t] = fsum[t - 155] * invN;
    __syncthreads();
    if (t < HDIM) {
        float a = b1[t];
        for (int k = 0; k < 158; ++k) a += cin[k] * w1[k * HDIM + t];
        mid[t] = a / (1.0f + expf(-a));
    }
    __syncthreads();
    if (t < HDIM) {
        float a = b2[t];
        for (int k = 0; k < HDIM; ++k) a += mid[k] * w2[k * HDIM + t];
        out[t] = a;
    }
}

// =====================================================================
extern "C" void kernel_launch(void* const* d_in, const int* in_sizes, int n_in,
                              void* d_out, int out_size, void* d_ws, size_t ws_size,
                              hipStream_t stream)
{
    const float* pos       = (const float*)d_in[0];
    const float* species   = (const float*)d_in[1];
    const float* forces    = (const float*)d_in[2];
    const int*   assign    = (const int*)  d_in[3];
    const float* lc        = (const float*)d_in[4];
    const float* gc        = (const float*)d_in[5];
    const float* a_w1      = (const float*)d_in[6];
    const float* a_b1      = (const float*)d_in[7];
    const float* a_w2      = (const float*)d_in[8];
    const float* a_b2      = (const float*)d_in[9];
    const float* g_w1      = (const float*)d_in[10];
    const float* g_b1      = (const float*)d_in[11];
    const float* g_w2      = (const float*)d_in[12];
    const float* g_b2      = (const float*)d_in[13];
    const float* c_w1      = (const float*)d_in[14];
    const float* c_b1      = (const float*)d_in[15];
    const float* c_w2      = (const float*)d_in[16];
    const float* c_b2      = (const float*)d_in[17];

    float* out = (float*)d_out;
    float* ws  = (float*)d_ws;

    // zero the accumulator region (group accums, counts, conformer sums)
    hipMemsetAsync((char*)d_ws + WS_GRP_ACC * sizeof(float), 0,
                   (WS_END - WS_GRP_ACC) * sizeof(float), stream);

    // --- stage 1: distances + RBFs + pooled sums -------------------------
    rbf_kernel<<<NATOMS, 256, 0, stream>>>(
        pos, species, forces, assign, lc, gc,
        out + OUT_LOCAL, out + OUT_GLOBAL, out + OUT_DIST,
        ws + WS_H, ws + WS_GRP_ACC, ws + WS_COUNTS,
        ws + WS_GPSUM, ws + WS_PSUM, ws + WS_FSUM);

    // --- stage 2: atomic MLP --------------------------------------------
    // layer 1: [1536,52]@[52,128] + b, SiLU -> mid1     (768 tiles / 8 waves)
    gemm_wmma_kernel<<<96, 256, 0, stream>>>(
        ws + WS_H, a_w1, a_b1, ws + WS_MID1,
        NATOMS, HDIM, 52, /*mode=*/0, nullptr, nullptr, nullptr);
    // layer 2: [1536,128]@[128,128] + b -> atomic out, + group/mean atomics
    gemm_wmma_kernel<<<96, 256, 0, stream>>>(
        ws + WS_MID1, a_w2, a_b2, out + OUT_ATOMIC,
        NATOMS, HDIM, HDIM, /*mode=*/2, assign,
        ws + WS_GRP_ACC, ws + WS_ASUM);

    // --- stage 3: group MLP ---------------------------------------------
    grp_finalize<<<(NGRP * 176 + 255) / 256, 256, 0, stream>>>(
        ws + WS_GRP_ACC, ws + WS_COUNTS, ws + WS_GRP_IN);
    // layer 1: [192,176]@[176,128] + b, SiLU -> gmid    (96 tiles / 8 waves)
    gemm_wmma_kernel<<<12, 256, 0, stream>>>(
        ws + WS_GRP_IN, g_w1, g_b1, ws + WS_GMID,
        NGRP, HDIM, 176, /*mode=*/0, nullptr, nullptr, nullptr);
    // layer 2: [192,128]@[128,128] + b -> group out
    gemm_wmma_kernel<<<12, 256, 0, stream>>>(
        ws + WS_GMID, g_w2, g_b2, out + OUT_GROUP,
        NGRP, HDIM, HDIM, /*mode=*/1, nullptr, nullptr, nullptr);

    // --- stage 4: conformer MLP (M=1) -----------------------------------
    conformer_kernel<<<1, 160, 0, stream>>>(
        ws + WS_ASUM, ws + WS_GPSUM, ws + WS_PSUM, ws + WS_FSUM,
        c_w1, c_b1, c_w2, c_b2, out + OUT_CONF);
}